// Logit_51908974739905
// MI455X (gfx1250) — compile-verified
//
#include <hip/hip_runtime.h>
#include <hip/hip_bf16.h>
#include <math.h>

typedef __attribute__((ext_vector_type(2))) float v2f;
typedef __attribute__((ext_vector_type(8))) float v8f;

#define DIM 128
#define WAVES_PER_BLOCK 8

// ---------------------------------------------------------------------------
// Phase 1: per batch, fold the complex Q projection and the K weights into a
// single complex vector u = conj(Wk)^T q'  plus a complex constant c = q'^T conj(bk).
// Writes, per batch:  w[0:128]=s*ur  w[128:256]=s*ui  w[256:384]=s*ui  w[384:512]=-s*ur
// (rows of the 2x256 folded "A matrix"), and cbuf[b*2..] = s*(cr, ci).
// ---------------------------------------------------------------------------
__global__ __launch_bounds__(128) void prep_kernel(
    const float* __restrict__ q_real, const float* __restrict__ q_imag,
    const float* __restrict__ Wq_r,  const float* __restrict__ Wq_i,
    const float* __restrict__ bq_r,  const float* __restrict__ bq_i,
    const float* __restrict__ Wk_r,  const float* __restrict__ Wk_i,
    const float* __restrict__ bk_r,  const float* __restrict__ bk_i,
    float* __restrict__ wbuf, float* __restrict__ cbuf)
{
    const int b = blockIdx.x;
    const int j = threadIdx.x;            // 0..127
    __shared__ float qin_r[DIM], qin_i[DIM], qpr[DIM], qpi[DIM], red[2 * DIM];

    qin_r[j] = q_real[b * DIM + j];
    qin_i[j] = q_imag[b * DIM + j];
    __syncthreads();

    // q' = complex linear of q with Wq (reference bias convention:
    // real gets +bq_r-bq_i, imag gets +bq_r+bq_i)
    float sr = 0.f, si = 0.f;
    for (int e = 0; e < DIM; ++e) {
        float wr = Wq_r[j * DIM + e], wi = Wq_i[j * DIM + e];
        sr = fmaf(qin_r[e], wr, sr); sr = fmaf(-qin_i[e], wi, sr);
        si = fmaf(qin_i[e], wr, si); si = fmaf( qin_r[e], wi, si);
    }
    float qr = sr + bq_r[j] - bq_i[j];
    float qi = si + bq_r[j] + bq_i[j];
    qpr[j] = qr; qpi[j] = qi;

    // constant term: c = q'^T conj(kb), kb = (bk_r - bk_i) + i(bk_r + bk_i)
    float kbr = bk_r[j] - bk_i[j];
    float kbi = bk_r[j] + bk_i[j];
    red[j]       = qr * kbr + qi * kbi;   // Re
    red[DIM + j] = qi * kbr - qr * kbi;   // Im
    __syncthreads();

    // u_d = sum_j q'_j * conj(Wk[j,d])
    const int d = threadIdx.x;
    float ur = 0.f, ui = 0.f;
    for (int jj = 0; jj < DIM; ++jj) {
        float wr = Wk_r[jj * DIM + d], wi = Wk_i[jj * DIM + d];
        ur = fmaf(qpr[jj], wr, ur); ur = fmaf( qpi[jj], wi, ur);
        ui = fmaf(qpi[jj], wr, ui); ui = fmaf(-qpr[jj], wi, ui);
    }
    const float s = 0.08838834764831845f; // 128^-0.5 (fold scaling in here)
    ur *= s; ui *= s;

    float* w = wbuf + b * 4 * DIM;
    w[d]           = ur;   // ar row, hr part
    w[DIM + d]     = ui;   // ar row, hi part
    w[2 * DIM + d] = ui;   // ai row, hr part
    w[3 * DIM + d] = -ur;  // ai row, hi part

    if (j == 0) {
        float c0 = 0.f, c1 = 0.f;
        for (int t = 0; t < DIM; ++t) { c0 += red[t]; c1 += red[DIM + t]; }
        cbuf[b * 2 + 0] = c0 * s;
        cbuf[b * 2 + 1] = c1 * s;
    }
}

// ---------------------------------------------------------------------------
// Phase 2: stream h (410 MB), 16 rows per wave. D(16x16) = A(16x256) x B(256x16)
// via 64 chained V_WMMA_F32_16X16X4_F32 (fp32, exact).
//   A: rows 0,1 = folded query vectors, rows 2-15 zero (staged in LDS in
//      fragment order -> conflict-free ds_load_b64 per chunk)
//   B: the 16 h-rows (feature dim = K), loaded as per-lane float2 straight
//      from global in the exact B-fragment layout.
//   D: acc[0]/acc[1] on lanes 0-15 = ar/ai per row -> epilogue inline.
// ---------------------------------------------------------------------------
__global__ __launch_bounds__(256) void attn_kernel(
    const float* __restrict__ h_real, const float* __restrict__ h_imag,
    const float* __restrict__ wbuf,   const float* __restrict__ cbuf,
    float* __restrict__ out, int N, int ntiles)
{
    const int b = blockIdx.y;

    // LDS A-matrix in fragment order: chunk kc occupies 64 contiguous floats,
    // lane l reads a float2 at  kc*64 + (l&15)*4 + (l>>4)*2   (all 32 lanes on
    // distinct bank pairs -> conflict free).
    __shared__ float Alds[64 * 64];  // 16 KB
    const float* w = wbuf + b * 512;
    for (int i = threadIdx.x; i < 64 * 64; i += 256) {
        int kc = i >> 6, within = i & 63;
        int m = within >> 2, kh = (within >> 1) & 1, jj = within & 1;
        int k = kc * 4 + kh * 2 + jj;              // 0..255 ([hr;hi] concat index)
        Alds[i] = (m < 2) ? w[m * 256 + k] : 0.0f; // rows >= 2 are zero
    }
    __syncthreads();

    const int wave = threadIdx.x >> 5;   // wave32
    const int lane = threadIdx.x & 31;
    const int tile = blockIdx.x * WAVES_PER_BLOCK + wave;
    if (tile >= ntiles) return;          // wave-uniform: EXEC stays all-ones

    const int mrow  = lane & 15;         // N index within tile / A row
    const int khalf = lane >> 4;         // K half select
    const int r  = tile * 16 + mrow;
    const int rr = (r < N) ? r : (N - 1);

    const size_t hoff = (size_t)b * N * DIM + (size_t)rr * DIM + khalf * 2;
    const float* hr = h_real + hoff;
    const float* hi = h_imag + hoff;
    const float* ap = Alds + mrow * 4 + khalf * 2;

    v8f acc = {};
#pragma unroll
    for (int kc = 0; kc < 32; ++kc) {    // K = 0..127  (h_real part)
        v2f a  = *(const v2f*)(ap + kc * 64);
        v2f bb = *(const v2f*)(hr + kc * 4);
        acc = __builtin_amdgcn_wmma_f32_16x16x4_f32(
                  false, a, false, bb, (short)0, acc, false, false);
    }
#pragma unroll
    for (int kc = 0; kc < 32; ++kc) {    // K = 128..255 (h_imag part)
        v2f a  = *(const v2f*)(ap + (kc + 32) * 64);
        v2f bb = *(const v2f*)(hi + kc * 4);
        acc = __builtin_amdgcn_wmma_f32_16x16x4_f32(
                  false, a, false, bb, (short)0, acc, false, false);
    }

    // Epilogue: lanes 0-15 hold (M=0 -> ar, M=1 -> ai) for row r.
    if (lane < 16 && r < N) {
        float ar = acc[0] + cbuf[b * 2 + 0];
        float ai = acc[1] + cbuf[b * 2 + 1];
        float mag   = sqrtf(fmaf(ar, ar, ai * ai));
        float scale = 10.0f * tanhf(mag) / fmaxf(mag, 1e-12f);
        float cr = scale * ar;
        float ci = -scale * ai;
        float sr = 1.0f / (1.0f + __expf(-cr));
        float si = 1.0f / (1.0f + __expf(-ci));
        float p  = fmaf(sr, sr, si * si);
        out[(size_t)b * N + r] = fminf(p, 1.0f);
    }
}

// ---------------------------------------------------------------------------
extern "C" void kernel_launch(void* const* d_in, const int* in_sizes, int n_in,
                              void* d_out, int out_size, void* d_ws, size_t ws_size,
                              hipStream_t stream)
{
    const float* q_real = (const float*)d_in[0];
    const float* q_imag = (const float*)d_in[1];
    const float* h_real = (const float*)d_in[2];
    const float* h_imag = (const float*)d_in[3];
    const float* Wq_r   = (const float*)d_in[4];
    const float* Wq_i   = (const float*)d_in[5];
    const float* bq_r   = (const float*)d_in[6];
    const float* bq_i   = (const float*)d_in[7];
    const float* Wk_r   = (const float*)d_in[8];
    const float* Wk_i   = (const float*)d_in[9];
    const float* bk_r   = (const float*)d_in[10];
    const float* bk_i   = (const float*)d_in[11];

    const int B = in_sizes[0] / DIM;           // q_real is [B,1,128]
    const int N = in_sizes[2] / (B * DIM);     // h_real is [B,N,128]

    float* wbuf = (float*)d_ws;                // B * 512 floats
    float* cbuf = wbuf + (size_t)B * 512;      // B * 2 floats

    prep_kernel<<<B, 128, 0, stream>>>(q_real, q_imag, Wq_r, Wq_i, bq_r, bq_i,
                                       Wk_r, Wk_i, bk_r, bk_i, wbuf, cbuf);

    const int ntiles = (N + 15) / 16;
    dim3 grid((ntiles + WAVES_PER_BLOCK - 1) / WAVES_PER_BLOCK, B);
    attn_kernel<<<grid, 256, 0, stream>>>(h_real, h_imag, wbuf, cbuf,
                                          (float*)d_out, N, ntiles);
}